// NMS3d_25056839205936
// MI455X (gfx1250) — compile-verified
//
#include <hip/hip_runtime.h>
#include <cstdint>

namespace {
constexpr int kH = 1536, kW = 1536, kD = 3;
constexpr int kHW = kH * kW;

constexpr int TW = 128;          // tile width
constexpr int TH = 8;            // tile height
constexpr int HALO_W = TW + 2;   // 130
constexpr int HALO_H = TH + 2;   // 10
constexpr int LSTR = HALO_W + 2; // 132, padded LDS row stride (bank stagger)
constexpr int NTHREADS = 256;
constexpr float kEPS = 1e-5f;
} // namespace

// gfx1250 async copy: global memory -> LDS, tracked by ASYNCcnt.
__device__ __forceinline__ void async_load_to_lds_b32(uint32_t lds_byte_off,
                                                      const float* src) {
  asm volatile("global_load_async_to_lds_b32 %0, %1, off"
               :
               : "v"(lds_byte_off), "v"(src)
               : "memory");
}

__device__ __forceinline__ void wait_async0() {
  asm volatile("s_wait_asynccnt 0x0" ::: "memory");
}

// Generic pointer to an LDS object: low 32 bits are the LDS byte offset.
__device__ __forceinline__ uint32_t lds_off(const void* p) {
  return (uint32_t)(uintptr_t)p;
}

__global__ __launch_bounds__(NTHREADS) void nms3d_kernel(
    const float* __restrict__ x, float* __restrict__ out) {
  __shared__ float s_x[kD][HALO_H][LSTR]; // 3 depth planes, with halo
  __shared__ float s_z[HALO_H][LSTR];     // depth-max plane, with halo

  const int tid = threadIdx.x;
  const int bx = blockIdx.x * TW;
  const int by = blockIdx.y * TH;
  const float* xb = x + (size_t)blockIdx.z * kD * kHW;
  float* ob = out + (size_t)blockIdx.z * kD * kHW;

  // ---- Phase 1: async-load 3 depth planes (edge-clamped halo) into LDS ----
  for (int i = tid; i < HALO_H * HALO_W; i += NTHREADS) {
    const int ly = i / HALO_W;
    const int lx = i - ly * HALO_W;
    const int gy = by + ly - 1;
    const int gx = bx + lx - 1;
    const int gyc = min(max(gy, 0), kH - 1);
    const int gxc = min(max(gx, 0), kW - 1);
    const float* src = xb + gyc * kW + gxc;
    async_load_to_lds_b32(lds_off(&s_x[0][ly][lx]), src);
    async_load_to_lds_b32(lds_off(&s_x[1][ly][lx]), src + kHW);
    async_load_to_lds_b32(lds_off(&s_x[2][ly][lx]), src + 2 * kHW);
  }
  wait_async0();
  __syncthreads();

  // ---- Phase 2: depth-max; -inf outside the image (pool pad is -inf) ----
  for (int i = tid; i < HALO_H * HALO_W; i += NTHREADS) {
    const int ly = i / HALO_W;
    const int lx = i - ly * HALO_W;
    const int gy = by + ly - 1;
    const int gx = bx + lx - 1;
    const float z =
        fmaxf(fmaxf(s_x[0][ly][lx], s_x[1][ly][lx]), s_x[2][ly][lx]);
    const bool oob = (gy < 0) | (gy >= kH) | (gx < 0) | (gx >= kW);
    s_z[ly][lx] = oob ? -__builtin_inff() : z;
  }
  __syncthreads();

  // ---- Phase 3: 3x3 spatial max, threshold, non-temporal store ----
  for (int i = tid; i < TH * TW; i += NTHREADS) {
    const int lx = i & (TW - 1);
    const int ly = i >> 7; // TW == 128
    const int hy = ly + 1;
    const int hx = lx + 1;

    float mp = s_z[hy - 1][hx - 1];
    mp = fmaxf(mp, s_z[hy - 1][hx]);
    mp = fmaxf(mp, s_z[hy - 1][hx + 1]);
    mp = fmaxf(mp, s_z[hy][hx - 1]);
    mp = fmaxf(mp, s_z[hy][hx]);
    mp = fmaxf(mp, s_z[hy][hx + 1]);
    mp = fmaxf(mp, s_z[hy + 1][hx - 1]);
    mp = fmaxf(mp, s_z[hy + 1][hx]);
    mp = fmaxf(mp, s_z[hy + 1][hx + 1]);

    const int gidx = (by + ly) * kW + (bx + lx);
#pragma unroll
    for (int p = 0; p < kD; ++p) {
      const float xv = s_x[p][hy][hx];
      const float o = (xv - mp + kEPS > 0.0f) ? xv : 0.0f;
      __builtin_nontemporal_store(o, &ob[p * kHW + gidx]);
    }
  }
}

extern "C" void kernel_launch(void* const* d_in, const int* in_sizes, int n_in,
                              void* d_out, int out_size, void* d_ws,
                              size_t ws_size, hipStream_t stream) {
  (void)n_in;
  (void)out_size;
  (void)d_ws;
  (void)ws_size;
  const float* x = (const float*)d_in[0];
  float* out = (float*)d_out;

  const int B = in_sizes[0] / (kD * kHW); // 8 for the reference shapes
  dim3 grid(kW / TW, kH / TH, B);         // (12, 192, 8)
  nms3d_kernel<<<grid, dim3(NTHREADS), 0, stream>>>(x, out);
}